// MAFLayer_62895501083197
// MI455X (gfx1250) — compile-verified
//
#include <hip/hip_runtime.h>
#include <math.h>

// ---------------- problem constants ----------------
#define BATCH   131072
#define NOUT    512          // OUTPUT_DIM
#define NIN     64           // INPUT_DIM
#define DTOT    576          // NIN + NOUT (K dimension)

// ---------------- tiling ----------------
#define MBLK    128          // rows per block
#define NBLK    256          // output cols per block  (grid.y = 2)
#define KC      16           // K-chunk streamed through LDS
#define K2C     (KC/2)       // float-pairs per chunk = 8
#define NCHUNK  (DTOT/KC)    // 36
#define THREADS 512          // 16 waves
#define XS_STRIDE 18         // 16 cols + 2 pad (8B align kept, banks spread)
#define XS_ELEMS (MBLK * XS_STRIDE)      // 2304 floats (9216 B)
#define WS_ELEMS (K2C * NBLK)            // 2048 float4 (32768 B)
#define SMEM_BYTES (2 * (WS_ELEMS * 16 + XS_ELEMS * 4))   // 83968 B double-buffered

typedef float v2f __attribute__((ext_vector_type(2)));
typedef float v8f __attribute__((ext_vector_type(8)));

#if defined(__AMDGCN__) && __has_builtin(__builtin_amdgcn_global_load_async_to_lds_b128)
#define MAF_ASYNC 1
#else
#define MAF_ASYNC 0
#endif

#if MAF_ASYNC
// exact parameter types per hipcc diagnostic:
//   param0: __attribute__((vector_size(16))) int  __device__(AS1) *
//   param1: (by symmetry) same vector type in AS3 (__shared__)
typedef int maf_v4i __attribute__((__vector_size__(4 * sizeof(int))));
typedef __attribute__((address_space(1))) maf_v4i* maf_gv4p;
typedef __attribute__((address_space(3))) maf_v4i* maf_lv4p;
#endif

__device__ __forceinline__ void maf_wait_asynccnt0() {
#if defined(__AMDGCN__)
#if __has_builtin(__builtin_amdgcn_s_wait_asynccnt)
    __builtin_amdgcn_s_wait_asynccnt(0);
#else
    asm volatile("s_wait_asynccnt 0x0" ::: "memory");
#endif
#endif
}

// ---------------------------------------------------------------------------
// Pack kernel: apply causal mask, transpose to k-major, interleave PA/PB pairs
// W4[k2*512 + n] = { PA[n][2k2]*m, PA[n][2k2+1]*m, PB[n][2k2]*m, PB[n][2k2+1]*m }
// ---------------------------------------------------------------------------
__global__ void maf_pack_weights(const float* __restrict__ PA,
                                 const float* __restrict__ PB,
                                 float4* __restrict__ W4)
{
    int id = blockIdx.x * 256 + threadIdx.x;          // [0, 288*512)
    if (id >= (DTOT / 2) * NOUT) return;
    int k2 = id >> 9;            // / 512
    int n  = id & (NOUT - 1);    // % 512
    int k  = 2 * k2;
    int lim = NIN + n;           // output col n uses k in [0, 64+n)
    float pa0 = (k     < lim) ? PA[(size_t)n * DTOT + k]     : 0.0f;
    float pa1 = (k + 1 < lim) ? PA[(size_t)n * DTOT + k + 1] : 0.0f;
    float pb0 = (k     < lim) ? PB[(size_t)n * DTOT + k]     : 0.0f;
    float pb1 = (k + 1 < lim) ? PB[(size_t)n * DTOT + k + 1] : 0.0f;
    W4[id] = make_float4(pa0, pa1, pb0, pb1);
}

// ---------------------------------------------------------------------------
// One K-chunk of dual-matrix WMMA accumulation (4 K-steps of K=4).
// ---------------------------------------------------------------------------
__device__ __forceinline__ void maf_compute_chunk(
    const float* __restrict__ xsb, const float4* __restrict__ wsb,
    int wm, int wn, int l16, int half,
    v8f accA[2][4], v8f accB[2][4])
{
#pragma unroll
    for (int ks = 0; ks < KC / 4; ++ks) {
        v2f a[2];
#pragma unroll
        for (int mt = 0; mt < 2; ++mt) {
            const float* p = &xsb[(wm * 32 + mt * 16 + l16) * XS_STRIDE
                                  + ks * 4 + 2 * half];
            a[mt][0] = p[0]; a[mt][1] = p[1];
        }
#pragma unroll
        for (int nt = 0; nt < 4; ++nt) {
            float4 w = wsb[(ks * 2 + half) * NBLK + wn * 64 + nt * 16 + l16];
            v2f bA; bA[0] = w.x; bA[1] = w.y;
            v2f bB; bB[0] = w.z; bB[1] = w.w;
#pragma unroll
            for (int mt = 0; mt < 2; ++mt) {
                accA[mt][nt] = __builtin_amdgcn_wmma_f32_16x16x4_f32(
                    false, a[mt], false, bA, (short)0, accA[mt][nt], false, false);
                accB[mt][nt] = __builtin_amdgcn_wmma_f32_16x16x4_f32(
                    false, a[mt], false, bB, (short)0, accB[mt][nt], false, false);
            }
        }
    }
}

// ---------------------------------------------------------------------------
// Main GEMM+epilogue kernel.
// Block: 512 threads = 16 waves arranged 4(M-groups of 32) x 4(N-groups of 64).
// Wave tile: M=32 (2 m-tiles) x N=64 (4 n-tiles), dual accumulators (A and Bv).
// K streamed in chunks of 16 through double-buffered LDS; next chunk's
// global->LDS traffic overlaps the current chunk's WMMAs (async-to-LDS when
// available, register-staged prefetch otherwise).
// ---------------------------------------------------------------------------
__global__ __launch_bounds__(THREADS)
void maf_main(const float* __restrict__ X,       // [B,512]
              const float* __restrict__ C,       // [B,64]
              const float4* __restrict__ W4,     // [288][512] packed masked wts
              float* __restrict__ Z,             // [B,512]
              float* __restrict__ LDPART)        // [2][B] per-(N-block) partials
{
    extern __shared__ float smem[];
    float4* wsb[2]; float* xsb[2];
    wsb[0] = (float4*)smem;
    wsb[1] = wsb[0] + WS_ELEMS;
    xsb[0] = (float*)(wsb[1] + WS_ELEMS);
    xsb[1] = xsb[0] + XS_ELEMS;

    const int tid  = threadIdx.x;
    const int lane = tid & 31;
    const int wave = tid >> 5;       // 0..15
    const int wm   = wave & 3;       // M group (32 rows each)
    const int wn   = wave >> 2;      // N group (64 cols each)
    const int l16  = lane & 15;
    const int half = lane >> 4;

    const int m0 = blockIdx.x * MBLK;
    const int n0 = blockIdx.y * NBLK;

    // per-thread transfer coordinates (hoisted)
    const int xr  = tid >> 2;            // x-tile row 0..127
    const int xc4 = (tid & 3) << 2;      // x-tile col 0,4,8,12
    float* xdst_base0 = &xsb[0][xr * XS_STRIDE + xc4];
    float* xdst_base1 = &xsb[1][xr * XS_STRIDE + xc4];

    v8f accA[2][4], accB[2][4];
#pragma unroll
    for (int mt = 0; mt < 2; ++mt)
#pragma unroll
        for (int nt = 0; nt < 4; ++nt)
#pragma unroll
            for (int e = 0; e < 8; ++e) { accA[mt][nt][e] = 0.0f; accB[mt][nt][e] = 0.0f; }

#if MAF_ASYNC
    // ------------------ async-to-LDS double-buffered pipeline ------------------
#define MAF_ISSUE(kcc, buf)                                                        \
    {                                                                              \
        const int kg  = (kcc) * KC;                                                \
        const float* gx = (kg < NIN)                                               \
            ? &C[(size_t)(m0 + xr) * NIN + kg + xc4]                               \
            : &X[(size_t)(m0 + xr) * NOUT + (kg - NIN) + xc4];                     \
        __builtin_amdgcn_global_load_async_to_lds_b128(                            \
            (maf_gv4p)(void*)(gx),                                                 \
            (maf_lv4p)(void*)((buf) ? xdst_base1 : xdst_base0), 0, 0);             \
        const int k2g = kg >> 1;                                                   \
        _Pragma("unroll")                                                          \
        for (int i = 0; i < 4; ++i) {                                              \
            int idx = tid + i * THREADS;                                           \
            int k2l = idx >> 8;                                                    \
            int nl  = idx & 255;                                                   \
            __builtin_amdgcn_global_load_async_to_lds_b128(                        \
                (maf_gv4p)(void*)(&W4[(size_t)(k2g + k2l) * NOUT + n0 + nl]),      \
                (maf_lv4p)(void*)(&wsb[(buf)][idx]), 0, 0);                        \
        }                                                                          \
    }

    MAF_ISSUE(0, 0);
    maf_wait_asynccnt0();
    __syncthreads();

    for (int kc = 0; kc < NCHUNK; ++kc) {
        const int cur = kc & 1;
        if (kc + 1 < NCHUNK) MAF_ISSUE(kc + 1, cur ^ 1);   // overlaps compute
        maf_compute_chunk(xsb[cur], wsb[cur], wm, wn, l16, half, accA, accB);
        maf_wait_asynccnt0();        // next buffer's async writes landed
        __syncthreads();             // publish to all waves / retire reads
    }
#undef MAF_ISSUE
#else
    // ------------- fallback: register-staged prefetch double buffer -------------
    // prologue: chunk 0 -> LDS buffer 0
    {
        float4 v = *(const float4*)&C[(size_t)(m0 + xr) * NIN + xc4]; // kg=0 < NIN
        xdst_base0[0] = v.x; xdst_base0[1] = v.y; xdst_base0[2] = v.z; xdst_base0[3] = v.w;
#pragma unroll
        for (int i = 0; i < 4; ++i) {
            int idx = tid + i * THREADS;
            int k2l = idx >> 8, nl = idx & 255;
            wsb[0][idx] = W4[(size_t)k2l * NOUT + n0 + nl];
        }
    }
    __syncthreads();

    for (int kc = 0; kc < NCHUNK; ++kc) {
        const int cur = kc & 1;
        float4 px; float4 pw[4];
        const bool have = (kc + 1 < NCHUNK);
        if (have) {                                  // global loads overlap compute
            const int kg = (kc + 1) * KC;
            px = (kg < NIN)
               ? *(const float4*)&C[(size_t)(m0 + xr) * NIN + kg + xc4]
               : *(const float4*)&X[(size_t)(m0 + xr) * NOUT + (kg - NIN) + xc4];
            const int k2g = kg >> 1;
#pragma unroll
            for (int i = 0; i < 4; ++i) {
                int idx = tid + i * THREADS;
                int k2l = idx >> 8, nl = idx & 255;
                pw[i] = W4[(size_t)(k2g + k2l) * NOUT + n0 + nl];
            }
        }
        maf_compute_chunk(xsb[cur], wsb[cur], wm, wn, l16, half, accA, accB);
        if (have) {
            float* xd = cur ? xdst_base0 : xdst_base1;   // write the other buffer
            xd[0] = px.x; xd[1] = px.y; xd[2] = px.z; xd[3] = px.w;
#pragma unroll
            for (int i = 0; i < 4; ++i)
                wsb[cur ^ 1][tid + i * THREADS] = pw[i];
        }
        __syncthreads();
    }
#endif

    // ---- epilogue: z = X*exp(A)+Bv, and deterministic row-sum of A ----
    // C/D layout: lanes 0-15 hold rows v (VGPR v), lanes 16-31 rows v+8; col = l16.
    float* part = smem;   // reuse LDS: [4 wn][128 rows] partial row sums
#pragma unroll
    for (int mt = 0; mt < 2; ++mt) {
#pragma unroll
        for (int v = 0; v < 8; ++v) {
            float s = accA[mt][0][v] + accA[mt][1][v] + accA[mt][2][v] + accA[mt][3][v];
            s += __shfl_xor(s, 1, 32);
            s += __shfl_xor(s, 2, 32);
            s += __shfl_xor(s, 4, 32);
            s += __shfl_xor(s, 8, 32);     // 16-lane group sum
            if (l16 == v)                  // lanes v and 16+v write unique slots
                part[wn * MBLK + wm * 32 + mt * 16 + v + half * 8] = s;
        }
#pragma unroll
        for (int nt = 0; nt < 4; ++nt) {
#pragma unroll
            for (int v = 0; v < 8; ++v) {
                int row = m0 + wm * 32 + mt * 16 + v + half * 8;
                int col = n0 + wn * 64 + nt * 16 + l16;
                size_t idx = (size_t)row * NOUT + col;
                Z[idx] = X[idx] * __expf(accA[mt][nt][v]) + accB[mt][nt][v];
            }
        }
    }
    __syncthreads();
    if (tid < MBLK) {
        float p = part[0 * MBLK + tid] + part[1 * MBLK + tid]
                + part[2 * MBLK + tid] + part[3 * MBLK + tid];
        LDPART[(size_t)blockIdx.y * BATCH + m0 + tid] = p;
    }
}

// final deterministic reduction of the two N-block partials
__global__ void maf_logdet_sum(const float* __restrict__ LDPART,
                               float* __restrict__ LOGDET)
{
    int i = blockIdx.x * 256 + threadIdx.x;
    if (i < BATCH)
        LOGDET[i] = LDPART[i] + LDPART[(size_t)BATCH + i];
}

// ---------------------------------------------------------------------------
extern "C" void kernel_launch(void* const* d_in, const int* in_sizes, int n_in,
                              void* d_out, int out_size, void* d_ws, size_t ws_size,
                              hipStream_t stream)
{
    const float* X  = (const float*)d_in[0];
    const float* C  = (const float*)d_in[1];
    const float* PA = (const float*)d_in[2];
    const float* PB = (const float*)d_in[3];

    float* Z      = (float*)d_out;                       // [B*512]
    float* LOGDET = Z + (size_t)BATCH * NOUT;            // [B]

    // workspace layout: packed weights (2.25 MB) then log_det partials (1 MB)
    float4* W4     = (float4*)d_ws;                                  // 288*512 float4
    float*  LDPART = (float*)((char*)d_ws + (size_t)(DTOT / 2) * NOUT * sizeof(float4));

    int packN = (DTOT / 2) * NOUT;
    maf_pack_weights<<<(packN + 255) / 256, 256, 0, stream>>>(PA, PB, W4);

    dim3 grid(BATCH / MBLK, NOUT / NBLK);                // (1024, 2)
    maf_main<<<grid, THREADS, SMEM_BYTES, stream>>>(X, C, W4, Z, LDPART);

    maf_logdet_sum<<<BATCH / 256, 256, 0, stream>>>(LDPART, LOGDET);
}